// GraphConvolution_73890617360949
// MI455X (gfx1250) — compile-verified
//
#include <hip/hip_runtime.h>

typedef __attribute__((ext_vector_type(2))) float v2f;
typedef __attribute__((ext_vector_type(8))) float v8f;

#define DIM 64

// ---------------------------------------------------------------------------
// Dense GEMM  XW = X @ W  with X:[M,64], W:[64,64], using fp32 WMMA.
// One wave (32 lanes) computes one 16x64 output tile:
//   - 4 accumulators of v8f (one per 16-col N tile)
//   - K loop over 64 in steps of 4 using V_WMMA_F32_16X16X4_F32
// A layout (16x4 f32): lanes 0-15 hold M=lane, K=k..k+1; lanes 16-31 hold
// M=lane-16, K=k+2..k+3  (ISA 7.12.2).
// B layout (4x16 f32): VGPR0/1 hold rows K (split by lane half), N=lane%16.
// C/D layout (16x16 f32): VGPR v, lanes 0-15 -> M=v, lanes 16-31 -> M=v+8.
// Grid is exact (M % 16 == 0 for 100000 and 50000), so EXEC is always all 1s.
// ---------------------------------------------------------------------------
__global__ __launch_bounds__(32) void gemm16xD_wmma(const float* __restrict__ X,
                                                    const float* __restrict__ W,
                                                    float* __restrict__ XW) {
  const int lane = threadIdx.x;      // 0..31
  const int m0   = blockIdx.x * 16;
  const int half = lane >> 4;        // 0 or 1 (K split)
  const int l16  = lane & 15;

  v8f acc[4] = {};
  const float* xrow = X + (size_t)(m0 + l16) * DIM;

#pragma unroll 4
  for (int k = 0; k < DIM; k += 4) {
    // A fragment: two consecutive f32 along K (8B aligned -> one b64 load)
    v2f a = *(const v2f*)(xrow + k + 2 * half);
#pragma unroll
    for (int nt = 0; nt < 4; ++nt) {
      v2f b;
      b.x = W[(size_t)(k + 2 * half)     * DIM + nt * 16 + l16];
      b.y = W[(size_t)(k + 2 * half + 1) * DIM + nt * 16 + l16];
      acc[nt] = __builtin_amdgcn_wmma_f32_16x16x4_f32(
          /*neg_a=*/false, a, /*neg_b=*/false, b,
          /*c_mod=*/(short)0, acc[nt], /*reuse_a=*/false, /*reuse_b=*/false);
    }
  }

#pragma unroll
  for (int nt = 0; nt < 4; ++nt) {
#pragma unroll
    for (int v = 0; v < 8; ++v) {
      XW[(size_t)(m0 + v + 8 * half) * DIM + nt * 16 + l16] = acc[nt][v];
    }
  }
}

// ---------------------------------------------------------------------------
// Zero-init output accumulators (float4 vectorized; total is divisible by 4).
// ---------------------------------------------------------------------------
__global__ __launch_bounds__(256) void zero_kernel(float4* __restrict__ p,
                                                   long long n4) {
  long long i = (long long)blockIdx.x * 256 + threadIdx.x;
  if (i < n4) {
    float4 z = {0.0f, 0.0f, 0.0f, 0.0f};
    p[i] = z;
  }
}

// ---------------------------------------------------------------------------
// COO SpMM, both directions, via L2-resident f32 atomics.
// 64 threads per edge: thread d handles feature dim d (coalesced 256B rows).
//   out_user[r, d] += v * xw_item[c, d]
//   out_item[c, d] += v * xw_user[r, d]
// ---------------------------------------------------------------------------
__global__ __launch_bounds__(256) void spmm_kernel(
    const int* __restrict__ rows, const int* __restrict__ cols,
    const float* __restrict__ vals,
    const float* __restrict__ xw_user, const float* __restrict__ xw_item,
    float* __restrict__ out_user, float* __restrict__ out_item, int nnz) {
  long long g = (long long)blockIdx.x * 256 + threadIdx.x;
  int e = (int)(g >> 6);
  int d = (int)(g & 63);
  if (e >= nnz) return;

  const int   r = rows[e];
  const int   c = cols[e];
  const float v = vals[e];

  const float xi = xw_item[(size_t)c * DIM + d];
  const float xu = xw_user[(size_t)r * DIM + d];
  atomicAdd(out_user + (size_t)r * DIM + d, v * xi);
  atomicAdd(out_item + (size_t)c * DIM + d, v * xu);
}

// ---------------------------------------------------------------------------
// In-place ReLU over the concatenated output (float4 vectorized).
// ---------------------------------------------------------------------------
__global__ __launch_bounds__(256) void relu_kernel(float4* __restrict__ p,
                                                   long long n4) {
  long long i = (long long)blockIdx.x * 256 + threadIdx.x;
  if (i < n4) {
    float4 v = p[i];
    v.x = fmaxf(v.x, 0.0f);
    v.y = fmaxf(v.y, 0.0f);
    v.z = fmaxf(v.z, 0.0f);
    v.w = fmaxf(v.w, 0.0f);
    p[i] = v;
  }
}

extern "C" void kernel_launch(void* const* d_in, const int* in_sizes, int n_in,
                              void* d_out, int out_size, void* d_ws, size_t ws_size,
                              hipStream_t stream) {
  const float* user_x  = (const float*)d_in[0];
  const float* item_x  = (const float*)d_in[1];
  const float* user_w  = (const float*)d_in[2];
  const float* item_w  = (const float*)d_in[3];
  const int*   ui_rows = (const int*)d_in[4];
  const int*   ui_cols = (const int*)d_in[5];
  const float* ui_vals = (const float*)d_in[6];

  const int n_users = in_sizes[0] / DIM;   // 100000
  const int n_items = in_sizes[1] / DIM;   // 50000
  const int nnz     = in_sizes[4];         // 1600000

  // Workspace: xw_user [n_users,64] then xw_item [n_items,64]
  float* xw_user = (float*)d_ws;
  float* xw_item = xw_user + (size_t)n_users * DIM;

  float* out_user = (float*)d_out;
  float* out_item = out_user + (size_t)n_users * DIM;

  const long long total = (long long)(n_users + n_items) * DIM;  // 9.6M
  const long long n4    = total / 4;

  // 1) zero the output accumulators
  zero_kernel<<<(unsigned)((n4 + 255) / 256), 256, 0, stream>>>(
      (float4*)d_out, n4);

  // 2) dense feature transforms via fp32 WMMA (one wave per 16-row tile)
  gemm16xD_wmma<<<n_users / 16, 32, 0, stream>>>(user_x, user_w, xw_user);
  gemm16xD_wmma<<<n_items / 16, 32, 0, stream>>>(item_x, item_w, xw_item);

  // 3) bidirectional COO SpMM with f32 atomics (L2-resident)
  const long long sp_threads = (long long)nnz * DIM;
  spmm_kernel<<<(unsigned)((sp_threads + 255) / 256), 256, 0, stream>>>(
      ui_rows, ui_cols, ui_vals, xw_user, xw_item, out_user, out_item, nnz);

  // 4) ReLU in place
  relu_kernel<<<(unsigned)((n4 + 255) / 256), 256, 0, stream>>>(
      (float4*)d_out, n4);
}